// CropRoi_18622978195957
// MI455X (gfx1250) — compile-verified
//
#include <hip/hip_runtime.h>
#include <stdint.h>

typedef __attribute__((ext_vector_type(2))) float v2f;
typedef __attribute__((ext_vector_type(8))) float v8f;

namespace {

constexpr int kSpat  = 2744;   // 14^3 spatial positions
constexpr int kTiles = 176;    // ceil(2744/16)=172, padded to multiple of 8 waves
constexpr int kWaves = 8;

// LDS layout (float offsets)
constexpr int L_FE1   = 0;        // 128*343 = 43904 : comb2 crop
constexpr int L_WUP   = 43904;    // 64*128  =  8192 : W_up2[cls]
constexpr int L_WBK   = 52096;    // 64*128  =  8192 : W_back2[cls]
constexpr int L_STAGE = 60288;    // 8*128*16 = 16384: per-wave GEMM2 input / stats scratch
constexpr int L_B1    = 76672;    // 64
constexpr int L_B2    = 76736;    // 64
constexpr int L_A1    = 76800;    // 64  scale1 = g1*rstd1
constexpr int L_C1    = 76864;    // 64  shift1 = (b1-mean1)*scale1 + be1
constexpr int L_A2    = 76928;    // 64
constexpr int L_C2    = 76992;    // 64
constexpr int L_TOTAL = 77056;    // 308224 bytes (<320KB WGP LDS)
constexpr int kSmemBytes = L_TOTAL * 4;

// align_corners=True, 7 -> 14: src = j*6/13
__device__ __forceinline__ void axis_interp(int j, int& i0, int& i1, float& w) {
  int p = j * 6;
  i0 = p / 13;
  w  = (float)(p - 13 * i0) * (1.0f / 13.0f);
  i1 = (i0 < 6) ? (i0 + 1) : 6;
}

__device__ __forceinline__ void tile_geom(int n0, int lane,
                                          bool& valid, int& jz, int& jy, int& jx,
                                          int off[8], float w[8]) {
  int coll = lane & 15;
  int colG = n0 + coll;
  valid = (colG < kSpat);
  int colC = valid ? colG : (kSpat - 1);
  jz = colC / 196;
  int r = colC - jz * 196;
  jy = r / 14;
  jx = r - jy * 14;
  int iz0, iz1, iy0, iy1, ix0, ix1;
  float wz, wy, wx;
  axis_interp(jz, iz0, iz1, wz);
  axis_interp(jy, iy0, iy1, wy);
  axis_interp(jx, ix0, ix1, wx);
  float wz0 = 1.f - wz, wy0 = 1.f - wy, wx0 = 1.f - wx;
  off[0] = iz0*49 + iy0*7 + ix0;  w[0] = wz0*wy0*wx0;
  off[1] = iz0*49 + iy0*7 + ix1;  w[1] = wz0*wy0*wx;
  off[2] = iz0*49 + iy1*7 + ix0;  w[2] = wz0*wy *wx0;
  off[3] = iz0*49 + iy1*7 + ix1;  w[3] = wz0*wy *wx;
  off[4] = iz1*49 + iy0*7 + ix0;  w[4] = wz *wy0*wx0;
  off[5] = iz1*49 + iy0*7 + ix1;  w[5] = wz *wy0*wx;
  off[6] = iz1*49 + iy1*7 + ix0;  w[6] = wz *wy *wx0;
  off[7] = iz1*49 + iy1*7 + ix1;  w[7] = wz *wy *wx;
}

__device__ __forceinline__ v8f wmma_f32(v2f a, v2f b, v8f c) {
  // D = A(16x4 f32) * B(4x16 f32) + C(16x16 f32)
  return __builtin_amdgcn_wmma_f32_16x16x4_f32(false, a, false, b, (short)0, c,
                                               false, false);
}

// acc[64x16] = W[64x128] * X[128x16]; X = trilinear-upsampled fe1 at this tile's
// 16 spatial columns. Each lane produces exactly its B-fragment elements.
__device__ __forceinline__ void gemm1(const float* __restrict__ fe1,
                                      const float* __restrict__ wlds,
                                      int lane, const int off[8], const float w[8],
                                      v8f acc[4]) {
  v8f z = {0.f,0.f,0.f,0.f,0.f,0.f,0.f,0.f};
  acc[0] = z; acc[1] = z; acc[2] = z; acc[3] = z;
  const int kbase = (lane < 16) ? 0 : 2;
  const int row0  = lane & 15;
  for (int kb = 0; kb < 32; ++kb) {
    const int k0 = kb * 4 + kbase;
    const float* f0 = fe1 + k0 * 343;
    float bx = 0.f, by = 0.f;
#pragma unroll
    for (int c = 0; c < 8; ++c) {
      bx = fmaf(w[c], f0[off[c]], bx);
      by = fmaf(w[c], f0[343 + off[c]], by);
    }
    v2f bf; bf.x = bx; bf.y = by;
#pragma unroll
    for (int m = 0; m < 4; ++m) {
      const v2f af = *(const v2f*)(wlds + (m * 16 + row0) * 128 + k0);
      acc[m] = wmma_f32(af, bf, acc[m]);
    }
  }
}

// acc[64x16] = W[64x128] * stage[128x16]
__device__ __forceinline__ void gemm2(const float* __restrict__ st,
                                      const float* __restrict__ wlds,
                                      int lane, v8f acc[4]) {
  v8f z = {0.f,0.f,0.f,0.f,0.f,0.f,0.f,0.f};
  acc[0] = z; acc[1] = z; acc[2] = z; acc[3] = z;
  const int kbase = (lane < 16) ? 0 : 2;
  const int row0  = lane & 15;
  for (int kb = 0; kb < 32; ++kb) {
    const int k0 = kb * 4 + kbase;
    v2f bf;
    bf.x = st[k0 * 16 + row0];
    bf.y = st[(k0 + 1) * 16 + row0];
#pragma unroll
    for (int m = 0; m < 4; ++m) {
      const v2f af = *(const v2f*)(wlds + (m * 16 + row0) * 128 + k0);
      acc[m] = wmma_f32(af, bf, acc[m]);
    }
  }
}

__device__ __forceinline__ void atomic_fmax(float* addr, float val) {
  asm volatile("global_atomic_max_num_f32 %0, %1, off"
               :: "v"((uint64_t)(uintptr_t)addr), "v"(val) : "memory");
}

// gemm1 -> norm1+relu into stage rows 0..63, o1 crop into rows 64..127, gemm2
__device__ __forceinline__ void compute_tile(const float* __restrict__ smem,
                                             const float* __restrict__ fe1,
                                             const float* __restrict__ wupL,
                                             const float* __restrict__ wbkL,
                                             float* __restrict__ stw,
                                             const float* __restrict__ out1,
                                             int b, int z2, int y2, int x2,
                                             int n0, int lane, int rowOff, int coll,
                                             const int off[8], const float w[8],
                                             v8f acc2[4]) {
  v8f acc[4];
  gemm1(fe1, wupL, lane, off, w, acc);
#pragma unroll
  for (int m = 0; m < 4; ++m)
#pragma unroll
    for (int r = 0; r < 8; ++r) {
      int row = m * 16 + r + rowOff;
      float v = fmaf(acc[m][r], smem[L_A1 + row], smem[L_C1 + row]);
      stw[row * 16 + coll] = fmaxf(v, 0.f);
    }
#pragma unroll 4
  for (int it = 0; it < 32; ++it) {
    int idx = lane + 32 * it;           // 0..1023 over [64 ch][16 col]
    int ch = idx >> 4;
    int cc = idx & 15;
    int cg = n0 + cc; cg = (cg < kSpat) ? cg : (kSpat - 1);
    int tz = cg / 196; int rr = cg - tz * 196;
    int ty = rr / 14;  int tx = rr - ty * 14;
    stw[(64 + ch) * 16 + cc] =
        out1[(((b * 64 + ch) * 64 + z2 + tz) * 64 + y2 + ty) * 64 + x2 + tx];
  }
  asm volatile("s_wait_dscnt 0x0" ::: "memory");
  gemm2(stw, wbkL, lane, acc2);
}

__global__ __launch_bounds__(256)
void crop_roi_wmma(const float* __restrict__ out1,
                   const float* __restrict__ comb2,
                   const int*   __restrict__ proposals,
                   const float* __restrict__ Wup,  const float* __restrict__ bup,
                   const float* __restrict__ gup,  const float* __restrict__ beup,
                   const float* __restrict__ Wbk,  const float* __restrict__ bbk,
                   const float* __restrict__ gbk,  const float* __restrict__ bebk,
                   const int*   __restrict__ cls_ind,
                   float*       __restrict__ out) {
  extern __shared__ float smem[];
  const int tid  = threadIdx.x;
  const int lane = tid & 31;
  const int wave = tid >> 5;
  const int p    = blockIdx.x;

  const int cls = cls_ind[0];
  const int b   = proposals[p * 7 + 0];
  const int z0  = proposals[p * 7 + 1];
  const int y0  = proposals[p * 7 + 2];
  const int x0  = proposals[p * 7 + 3];
  const int z4 = z0 >> 2, y4 = y0 >> 2, x4 = x0 >> 2;
  const int z2 = z0 >> 1, y2 = y0 >> 1, x2 = x0 >> 1;

  float* fe1   = smem + L_FE1;
  float* wupL  = smem + L_WUP;
  float* wbkL  = smem + L_WBK;
  float* stage = smem + L_STAGE;
  float* stw   = stage + wave * (128 * 16);

  // ---- stage comb2 crop [128,7,7,7] + weights + biases into LDS ----
  for (int i = tid; i < 128 * 343; i += 256) {
    int k = i / 343, r = i - k * 343;
    int z = r / 49;  r -= z * 49;
    int y = r / 7;
    int x = r - y * 7;
    fe1[i] = comb2[(((b * 128 + k) * 32 + z4 + z) * 32 + y4 + y) * 32 + x4 + x];
  }
  const float* WupG = Wup + cls * 64 * 128;
  const float* WbkG = Wbk + cls * 64 * 128;
  for (int i = tid; i < 64 * 128; i += 256) { wupL[i] = WupG[i]; wbkL[i] = WbkG[i]; }
  if (tid < 64) {
    smem[L_B1 + tid] = bup[cls * 64 + tid];
    smem[L_B2 + tid] = bbk[cls * 64 + tid];
  }
  __syncthreads();

  const int rowOff = (lane < 16) ? 0 : 8;
  const int coll   = lane & 15;

  // ===== phase 1: stats of conv1(up2(fe1)) + b1 =====
  {
    float s[4][8] = {}, q[4][8] = {};
    for (int t = wave; t < kTiles; t += kWaves) {
      bool valid; int jz, jy, jx, off[8]; float w[8];
      tile_geom(t * 16, lane, valid, jz, jy, jx, off, w);
      v8f acc[4];
      gemm1(fe1, wupL, lane, off, w, acc);
      const float msk = valid ? 1.f : 0.f;
#pragma unroll
      for (int m = 0; m < 4; ++m)
#pragma unroll
        for (int r = 0; r < 8; ++r) {
          float v = (acc[m][r] + smem[L_B1 + m * 16 + r + rowOff]) * msk;
          s[m][r] += v;
          q[m][r]  = fmaf(v, v, q[m][r]);
        }
    }
    float* red = stage + tid * 64;   // each lane's red slot lies in its own wave's stw
#pragma unroll
    for (int m = 0; m < 4; ++m)
#pragma unroll
      for (int r = 0; r < 8; ++r) { red[m*8+r] = s[m][r]; red[32+m*8+r] = q[m][r]; }
  }
  __syncthreads();
  if (tid < 64) {   // deterministic fixed-order reduction, one thread per channel
    int m = tid >> 4, rem = tid & 15, r = rem & 7, hi = rem >> 3;
    float sum = 0.f, sq = 0.f;
    for (int wv = 0; wv < kWaves; ++wv)
      for (int l = 0; l < 16; ++l) {
        const float* pp = stage + (wv * 32 + hi * 16 + l) * 64 + m * 8 + r;
        sum += pp[0]; sq += pp[32];
      }
    float mean = sum * (1.f / kSpat);
    float var  = sq  * (1.f / kSpat) - mean * mean;
    float rstd = rsqrtf(var + 1e-5f);
    float A = gup[cls * 64 + tid] * rstd;
    smem[L_A1 + tid] = A;
    smem[L_C1 + tid] = (smem[L_B1 + tid] - mean) * A + beup[cls * 64 + tid];
  }
  __syncthreads();

  // ===== phase 2: stats of conv2(cat(norm1relu, o1c)) + b2 =====
  {
    float s[4][8] = {}, q[4][8] = {};
    for (int t = wave; t < kTiles; t += kWaves) {
      bool valid; int jz, jy, jx, off[8]; float w[8];
      tile_geom(t * 16, lane, valid, jz, jy, jx, off, w);
      v8f acc2[4];
      compute_tile(smem, fe1, wupL, wbkL, stw, out1, b, z2, y2, x2,
                   t * 16, lane, rowOff, coll, off, w, acc2);
      const float msk = valid ? 1.f : 0.f;
#pragma unroll
      for (int m = 0; m < 4; ++m)
#pragma unroll
        for (int r = 0; r < 8; ++r) {
          float v = (acc2[m][r] + smem[L_B2 + m * 16 + r + rowOff]) * msk;
          s[m][r] += v;
          q[m][r]  = fmaf(v, v, q[m][r]);
        }
    }
    float* red = stage + tid * 64;
#pragma unroll
    for (int m = 0; m < 4; ++m)
#pragma unroll
      for (int r = 0; r < 8; ++r) { red[m*8+r] = s[m][r]; red[32+m*8+r] = q[m][r]; }
  }
  __syncthreads();
  if (tid < 64) {
    int m = tid >> 4, rem = tid & 15, r = rem & 7, hi = rem >> 3;
    float sum = 0.f, sq = 0.f;
    for (int wv = 0; wv < kWaves; ++wv)
      for (int l = 0; l < 16; ++l) {
        const float* pp = stage + (wv * 32 + hi * 16 + l) * 64 + m * 8 + r;
        sum += pp[0]; sq += pp[32];
      }
    float mean = sum * (1.f / kSpat);
    float var  = sq  * (1.f / kSpat) - mean * mean;
    float rstd = rsqrtf(var + 1e-5f);
    float A = gbk[cls * 64 + tid] * rstd;
    smem[L_A2 + tid] = A;
    smem[L_C2 + tid] = (smem[L_B2 + tid] - mean) * A + bebk[cls * 64 + tid];
  }
  __syncthreads();

  // ===== phase 3: final pass, norm2+relu, fused 2x2x2 maxpool via atomic fmax =====
  for (int t = wave; t < kTiles; t += kWaves) {
    bool valid; int jz, jy, jx, off[8]; float w[8];
    tile_geom(t * 16, lane, valid, jz, jy, jx, off, w);
    v8f acc2[4];
    compute_tile(smem, fe1, wupL, wbkL, stw, out1, b, z2, y2, x2,
                 t * 16, lane, rowOff, coll, off, w, acc2);
    const float msk = valid ? 1.f : 0.f;
    const int obase = p * 64 * 343 + (jz >> 1) * 49 + (jy >> 1) * 7 + (jx >> 1);
#pragma unroll
    for (int m = 0; m < 4; ++m)
#pragma unroll
      for (int r = 0; r < 8; ++r) {
        int row = m * 16 + r + rowOff;
        float v = fmaf(acc2[m][r], smem[L_A2 + row], smem[L_C2 + row]);
        v = fmaxf(v, 0.f) * msk;   // invalid cols contribute 0 == neutral for post-relu max
        atomic_fmax(out + obase + row * 343, v);
      }
  }
}

} // namespace

extern "C" void kernel_launch(void* const* d_in, const int* in_sizes, int n_in,
                              void* d_out, int out_size, void* d_ws, size_t ws_size,
                              hipStream_t stream) {
  const float* out1      = (const float*)d_in[1];
  const float* comb2     = (const float*)d_in[2];
  const int*   proposals = (const int*)d_in[4];
  const float* Wup  = (const float*)d_in[5];
  const float* bup  = (const float*)d_in[6];
  const float* gup  = (const float*)d_in[7];
  const float* beup = (const float*)d_in[8];
  const float* Wbk  = (const float*)d_in[9];
  const float* bbk  = (const float*)d_in[10];
  const float* gbk  = (const float*)d_in[11];
  const float* bebk = (const float*)d_in[12];
  const int*   cls  = (const int*)d_in[13];
  float* out = (float*)d_out;

  const int nprop = in_sizes[4] / 7;   // 512

  // post-relu values are >= 0, so zero-init + atomic fmax == exact maxpool
  (void)hipMemsetAsync(out, 0, (size_t)out_size * sizeof(float), stream);
  (void)hipFuncSetAttribute((const void*)crop_roi_wmma,
                            hipFuncAttributeMaxDynamicSharedMemorySize, kSmemBytes);
  crop_roi_wmma<<<dim3(nprop), dim3(256), kSmemBytes, stream>>>(
      out1, comb2, proposals, Wup, bup, gup, beup, Wbk, bbk, gbk, bebk, cls, out);
}